// PositionalEncoding_1580547967908
// MI455X (gfx1250) — compile-verified
//
#include <hip/hip_runtime.h>

typedef __attribute__((ext_vector_type(2))) float v2f;
typedef __attribute__((ext_vector_type(8))) float v8f;

#define SEQ   2048
#define HEADS 8
#define DK    64
#define NPOS  129      // max_pos + 1
#define PSTR  132      // padded LDS row stride (floats)

// One block handles (b, pair of seq rows i0, i0+1).
// Phase 1: 16x129 projection tile via v_wmma_f32_16x16x4_f32 into LDS.
//   Row mapping r = il*8 + h  (il = which of the two seq rows, h = head).
// Phase 2: gather with coalesced NT output stores.
__global__ __launch_bounds__(256) void rpr_fused_kernel(
    const float* __restrict__ q,     // [bs, 8, 2048, 64]
    const int*   __restrict__ dist,  // [bs, 2048, 2048]
    const float* __restrict__ w,     // [129, 64]
    float*       __restrict__ out)   // [bs, 8, 2048, 2048]
{
    __shared__ float proj[16 * PSTR];

    const int b   = blockIdx.y;
    const int i0  = blockIdx.x * 2;
    const int tid = threadIdx.x;
    const int lane = tid & 31;           // wave32
    const int wave = tid >> 5;           // 8 waves per block

    // ---------------- GEMM phase (all waves, WMMA f32 16x16x4) ----------------
    {
        const int row = lane & 15;       // M index 0..15
        const int hl  = lane >> 4;       // half-wave selects K sub-pair
        const int il  = row >> 3;        // which seq row of the pair
        const int h   = row & 7;         // head
        const float* qrow =
            q + (((size_t)b * HEADS + h) * SEQ + (size_t)(i0 + il)) * DK;

        // Preload A: a[kk] holds K = 4*kk + 2*hl + {0,1} of row `row`.
        v2f a[16];
        #pragma unroll
        for (int kk = 0; kk < 16; ++kk)
            a[kk] = *(const v2f*)(qrow + 4 * kk + 2 * hl);

        // 9 N-tiles of 16 cover p = 0..143 (cols >= 129 computed but discarded).
        for (int nt = wave; nt < 9; nt += 8) {
            const int p = nt * 16 + (lane & 15);              // N index
            const float* wrow = w + (size_t)((p < NPOS) ? p : 0) * DK; // clamp OOB reads
            v8f c = {};
            #pragma unroll
            for (int kk = 0; kk < 16; ++kk) {
                v2f bb = *(const v2f*)(wrow + 4 * kk + 2 * hl);
                c = __builtin_amdgcn_wmma_f32_16x16x4_f32(
                        /*neg_a=*/false, a[kk], /*neg_b=*/false, bb,
                        /*c_mod=*/(short)0, c, /*reuse_a=*/false, /*reuse_b=*/false);
            }
            // C layout: c[v] -> row (v + 8*(lane>>4)), col (lane&15) of the tile.
            if (p < NPOS) {
                #pragma unroll
                for (int v = 0; v < 8; ++v)
                    proj[(v + 8 * hl) * PSTR + p] = c[v];
            }
        }
    }

    __syncthreads();

    // ---------------- Gather phase (all 256 threads) ----------------
    const int* __restrict__ drow0 = dist + ((size_t)b * SEQ + i0) * SEQ;
    const int* __restrict__ drow1 = drow0 + SEQ;

    for (int j = tid; j < SEQ; j += 256) {
        int p0 = __builtin_nontemporal_load(drow0 + j);
        int p1 = __builtin_nontemporal_load(drow1 + j);
        p0 = (p0 > NPOS - 1) ? (NPOS - 1) : p0;
        p1 = (p1 > NPOS - 1) ? (NPOS - 1) : p1;
        #pragma unroll
        for (int h = 0; h < HEADS; ++h) {
            float v0 = proj[h * PSTR + p0];          // il = 0 -> row h
            float v1 = proj[(8 + h) * PSTR + p1];    // il = 1 -> row 8 + h
            float* o = out + (((size_t)(b * HEADS + h)) * SEQ + i0) * (size_t)SEQ + j;
            __builtin_nontemporal_store(v0, o);
            __builtin_nontemporal_store(v1, o + SEQ);
        }
    }
}

extern "C" void kernel_launch(void* const* d_in, const int* in_sizes, int n_in,
                              void* d_out, int out_size, void* d_ws, size_t ws_size,
                              hipStream_t stream) {
    const float* q    = (const float*)d_in[0];
    const int*   dist = (const int*)d_in[1];
    const float* w    = (const float*)d_in[2];
    float*       out  = (float*)d_out;

    const int bs = in_sizes[0] / (HEADS * SEQ * DK);   // 4 for the reference shapes
    dim3 grid(SEQ / 2, bs);
    rpr_fused_kernel<<<grid, 256, 0, stream>>>(q, dist, w, out);
}